// ScaledDotProductAttention_18030272708887
// MI455X (gfx1250) — compile-verified
//
#include <hip/hip_runtime.h>
#include <hip/hip_bf16.h>

// Shapes (fixed by the reference)
#define BB 4
#define HH 16
#define SS 2048
#define DD 64

#define ROWS 16          // q-rows per workgroup
#define LDS_STRIDE 2052  // 2048 + 4 floats pad -> conflict-free column access; keeps 16B align
#define NWAVES 4
#define NTHREADS 128
#define CHUNK 256        // columns per thread in softmax phase (S / 8)

typedef __attribute__((ext_vector_type(2))) float v2f;
typedef __attribute__((ext_vector_type(4))) float v4f;
typedef __attribute__((ext_vector_type(8))) float v8f;

__launch_bounds__(NTHREADS)
__global__ void sdpa_f32_wmma_kernel(const float* __restrict__ qg,
                                     const float* __restrict__ kg,
                                     const float* __restrict__ vg,
                                     const int*   __restrict__ maskg,
                                     float* __restrict__ outg,   // [B,H,S,D]
                                     float* __restrict__ attng)  // [B,H,S,S]
{
    extern __shared__ float smem[];
    float* sP  = smem;                          // ROWS * LDS_STRIDE scores/probs
    float* red = smem + ROWS * LDS_STRIDE;      // ROWS * 8 reduction scratch

    const int h  = blockIdx.x;   // heads fastest -> same-b blocks share mask via L2
    const int qt = blockIdx.y;
    const int b  = blockIdx.z;
    const int q0 = qt * ROWS;

    const int tid  = threadIdx.x;
    const int wave = tid >> 5;
    const int lane = tid & 31;
    const int nl   = lane & 15;   // column (B/C/D) or row (A) index within a tile
    const int hi   = lane >> 4;   // lane half selects K / M sub-block

    const size_t bh = (size_t)b * HH + h;
    const float* Q = qg + bh * (size_t)SS * DD;
    const float* K = kg + bh * (size_t)SS * DD;
    const float* V = vg + bh * (size_t)SS * DD;
    const int*   M = maskg + (size_t)b * SS * SS;

    const float scale = 0.125f;   // 1/sqrt(64)

    // ---------- load Q A-fragments once per wave: 16 chunks of K=4 ----------
    // A 16x4 fp32 layout: lane m = nl; VGPR0 = K = 2*hi, VGPR1 = K = 2*hi+1
    v2f aq[16];
    {
        const float* qrow = Q + (size_t)(q0 + nl) * DD + 2 * hi;
        #pragma unroll
        for (int kc = 0; kc < 16; ++kc) {
            aq[kc].x = qrow[4 * kc + 0];
            aq[kc].y = qrow[4 * kc + 1];
        }
    }

    // ---------- phase 1: scores = scale * (Q K^T), masked, into LDS ----------
    for (int nt = wave; nt < SS / 16; nt += NWAVES) {
        const int n0 = nt * 16;
        const float* krow = K + (size_t)(n0 + nl) * DD + 2 * hi;  // B: lane = N
        v8f acc = {};
        #pragma unroll
        for (int kc = 0; kc < 16; ++kc) {
            v2f bf;
            bf.x = krow[4 * kc + 0];
            bf.y = krow[4 * kc + 1];
            acc = __builtin_amdgcn_wmma_f32_16x16x4_f32(
                false, aq[kc], false, bf, (short)0, acc, false, false);
        }
        #pragma unroll
        for (int r = 0; r < 8; ++r) {
            const int row = r + 8 * hi;               // C/D layout
            float s = acc[r] * scale;
            const int mk = M[(size_t)(q0 + row) * SS + (n0 + nl)];
            s = (mk == 0) ? -1.0e9f : s;
            sP[row * LDS_STRIDE + n0 + nl] = s;
        }
    }
    __syncthreads();

    // ---------- phase 2: row softmax (float4 passes); stream attn with NT ----------
    {
        const int srow = tid & 15;          // 0..15
        const int sl   = tid >> 4;          // 0..7; each owns a contiguous 256-col chunk
        float* rowp = sP + srow * LDS_STRIDE + sl * CHUNK;   // 16B aligned

        float m = -3.0e38f;
        #pragma unroll 4
        for (int i = 0; i < CHUNK; i += 4) {
            const v4f x = *(const v4f*)(rowp + i);
            m = fmaxf(m, fmaxf(fmaxf(x.x, x.y), fmaxf(x.z, x.w)));
        }
        red[srow * 8 + sl] = m;
        __syncthreads();

        float rmax = red[srow * 8 + 0];
        #pragma unroll
        for (int j = 1; j < 8; ++j) rmax = fmaxf(rmax, red[srow * 8 + j]);
        __syncthreads();   // done reading red; safe to overwrite below

        float psum = 0.0f;
        #pragma unroll 4
        for (int i = 0; i < CHUNK; i += 4) {
            v4f x = *(const v4f*)(rowp + i);
            x.x = __expf(x.x - rmax);
            x.y = __expf(x.y - rmax);
            x.z = __expf(x.z - rmax);
            x.w = __expf(x.w - rmax);
            *(v4f*)(rowp + i) = x;
            psum += (x.x + x.y) + (x.z + x.w);
        }
        red[srow * 8 + sl] = psum;
        __syncthreads();

        float denom = 0.0f;
        #pragma unroll
        for (int j = 0; j < 8; ++j) denom += red[srow * 8 + j];
        const float inv = 1.0f / denom;

        // attn written exactly once, never re-read -> non-temporal (preserve L2
        // for the cross-head mask reuse and the K/V slabs).
        float* arow = attng + (bh * SS + (size_t)(q0 + srow)) * SS + sl * CHUNK;
        #pragma unroll 4
        for (int i = 0; i < CHUNK; i += 4) {
            v4f x = *(const v4f*)(rowp + i);
            x = x * inv;
            *(v4f*)(rowp + i) = x;     // keep normalized probs in LDS for phase 3
            __builtin_nontemporal_store(x, (v4f*)(arow + i));
        }
    }
    __syncthreads();

    // ---------- phase 3: out = P @ V, one 16-wide D slice per wave ----------
    {
        const int d0 = wave * 16;
        const float* vcol = V + d0 + nl;
        const float* arow = sP + nl * LDS_STRIDE + 2 * hi;   // A: lane = M row
        v8f oacc = {};
        #pragma unroll 4
        for (int k0 = 0; k0 < SS; k0 += 4) {
            v2f af;
            af.x = arow[k0 + 0];
            af.y = arow[k0 + 1];
            v2f bf;
            bf.x = vcol[(size_t)(k0 + 2 * hi + 0) * DD];
            bf.y = vcol[(size_t)(k0 + 2 * hi + 1) * DD];
            oacc = __builtin_amdgcn_wmma_f32_16x16x4_f32(
                false, af, false, bf, (short)0, oacc, false, false);
        }
        float* obase = outg + bh * (size_t)SS * DD;
        #pragma unroll
        for (int r = 0; r < 8; ++r) {
            const int row = q0 + r + 8 * hi;
            __builtin_nontemporal_store(oacc[r], obase + (size_t)row * DD + d0 + nl);
        }
    }
}

extern "C" void kernel_launch(void* const* d_in, const int* in_sizes, int n_in,
                              void* d_out, int out_size, void* d_ws, size_t ws_size,
                              hipStream_t stream) {
    (void)in_sizes; (void)n_in; (void)out_size; (void)d_ws; (void)ws_size;
    const float* q    = (const float*)d_in[0];
    const float* k    = (const float*)d_in[1];
    const float* v    = (const float*)d_in[2];
    const int*   mask = (const int*)d_in[3];

    float* out  = (float*)d_out;                               // [B,H,S,D] first
    float* attn = out + (size_t)BB * HH * SS * DD;             // then [B,H,S,S]

    // heads fastest in dispatch order -> mask rows for a given (b, q-tile)
    // are reused across all 16 heads through L2.
    dim3 grid(HH, SS / ROWS, BB);
    const size_t lds_bytes = (size_t)(ROWS * LDS_STRIDE + ROWS * 8) * sizeof(float);
    sdpa_f32_wmma_kernel<<<grid, NTHREADS, lds_bytes, stream>>>(q, k, v, mask, out, attn);
}